// ACNet_44702019616960
// MI455X (gfx1250) — compile-verified
//
#include <hip/hip_runtime.h>
#include <hip/hip_bf16.h>
#include <math.h>

// ---------------------------------------------------------------------------
// GAT 2-layer forward for MI455X (gfx1250, wave32).
//   layer1: 4-head GAT (128 -> 64 per head, concat 256) + ELU
//   layer2: 1-head GAT (256 -> 64)
// Dense projections use V_WMMA_F32_16X16X4_F32 (exact f32 matrix pipe).
// Edge softmax: 3 passes (atomic max via ordered-uint key, atomic sum, scatter).
// ---------------------------------------------------------------------------

typedef float v2f __attribute__((ext_vector_type(2)));
typedef float v8f __attribute__((ext_vector_type(8)));

#define IN_DIM  128
#define HID     64
#define HEADS   4
#define OUTD    64
#define NEG_SLOPE 0.01f

// ---------------- ordered float <-> uint key (monotone) --------------------
__device__ __forceinline__ unsigned enc_f(float x) {
    unsigned u = __float_as_uint(x);
    return (u >> 31) ? ~u : (u | 0x80000000u);
}
__device__ __forceinline__ float dec_f(unsigned k) {
    return (k >> 31) ? __uint_as_float(k & 0x7fffffffu) : __uint_as_float(~k);
}

// ---------------- WMMA f32 GEMM: D[rows x width] = A[rows x K] * B ----------
// One wave -> one 16x16 tile of D. B is [head][K][64] with row stride 64
// elements and head stride b_head_stride elements (head = gcol/64).
// A-frag layout (ISA 7.12.2, 32-bit A 16x4): lane&15 = M row,
//   lanes 0-15 hold K={k,k+1}, lanes 16-31 hold K={k+2,k+3}.
// D layout: VGPR v -> M = v + 8*(lane>>4), N = lane&15.
__global__ __launch_bounds__(256)
void gemm_wmma_f32(const float* __restrict__ A, int lda, int K,
                   const float* __restrict__ B, int b_head_stride,
                   float* __restrict__ D, int ldd,
                   int width, int n_rows) {
    const int lane = threadIdx.x & 31;
    const int wid  = blockIdx.x * (blockDim.x >> 5) + (threadIdx.x >> 5);
    const int col_tiles = width >> 4;
    const int mt = wid / col_tiles;
    const int ct = wid - mt * col_tiles;
    if (mt * 16 >= n_rows) return;            // wave-uniform -> EXEC stays all-1s

    const int l15  = lane & 15;
    const int kh   = lane >> 4;               // which K-pair this half-wave owns
    const int row  = mt * 16 + l15;
    const int gcol = ct * 16 + l15;
    const int head = gcol >> 6;               // 64 cols per head
    const int c    = gcol & 63;

    const float* Arow = A + (size_t)row * lda;
    const float* Bcol = B + (size_t)head * b_head_stride + c;

    v8f acc = {};
    for (int k = 0; k < K; k += 4) {
        const int kk = k + 2 * kh;
        v2f a, b;
        a[0] = Arow[kk];
        a[1] = Arow[kk + 1];
        b[0] = Bcol[(size_t)kk * 64];
        b[1] = Bcol[(size_t)(kk + 1) * 64];
        acc = __builtin_amdgcn_wmma_f32_16x16x4_f32(
            /*neg_a=*/false, a, /*neg_b=*/false, b,
            /*c_mod=*/(short)0, acc, /*reuse_a=*/false, /*reuse_b=*/false);
    }

    float* Dp = D + gcol;
    const int rbase = mt * 16 + 8 * kh;
#pragma unroll
    for (int v = 0; v < 8; ++v)
        Dp[(size_t)(rbase + v) * ldd] = acc[v];
}

// ---------------- per-(node,head) attention scores --------------------------
// s_src[n,h] = z[n,h,:] . a[h][0:dout] ; s_dst[n,h] = z[n,h,:] . a[h][dout:2dout]
__global__ __launch_bounds__(256)
void scores_kernel(const float* __restrict__ z, const float* __restrict__ a,
                   float* __restrict__ s_src, float* __restrict__ s_dst,
                   int n, int H, int dout) {
    int idx = blockIdx.x * blockDim.x + threadIdx.x;
    if (idx >= n * H) return;
    int node = idx / H, hh = idx - node * H;
    const float* zr = z + (size_t)(node * H + hh) * dout;
    const float* as = a + (size_t)hh * 2 * dout;
    const float* ad = as + dout;
    float ss = 0.f, sd = 0.f;
    for (int c = 0; c < dout; ++c) { float v = zr[c]; ss += v * as[c]; sd += v * ad[c]; }
    s_src[idx] = ss;
    s_dst[idx] = sd;
}

// ---------------- init max-keys to encoding of -inf -------------------------
__global__ void init_keys_kernel(unsigned* __restrict__ k, int n) {
    int i = blockIdx.x * blockDim.x + threadIdx.x;
    if (i < n) k[i] = enc_f(-INFINITY);
}

// ---------------- edge pass 1: segment max of leaky(logit) ------------------
__global__ __launch_bounds__(256)
void edge_max_kernel(const int* __restrict__ src, const int* __restrict__ dst,
                     const float* __restrict__ s_src, const float* __restrict__ s_dst,
                     unsigned* __restrict__ mkey, int E, int H) {
    int idx = blockIdx.x * blockDim.x + threadIdx.x;
    if (idx >= E * H) return;
    int e = idx / H, hh = idx - e * H;
    int s = src[e], d = dst[e];
    float lg = s_src[s * H + hh] + s_dst[d * H + hh];
    lg = lg >= 0.f ? lg : NEG_SLOPE * lg;
    atomicMax(&mkey[d * H + hh], enc_f(lg));
}

// ---------------- edge pass 2: segment sum of exp(logit - max) --------------
__global__ __launch_bounds__(256)
void edge_sum_kernel(const int* __restrict__ src, const int* __restrict__ dst,
                     const float* __restrict__ s_src, const float* __restrict__ s_dst,
                     const unsigned* __restrict__ mkey, float* __restrict__ ssum,
                     int E, int H) {
    int idx = blockIdx.x * blockDim.x + threadIdx.x;
    if (idx >= E * H) return;
    int e = idx / H, hh = idx - e * H;
    int s = src[e], d = dst[e];
    float lg = s_src[s * H + hh] + s_dst[d * H + hh];
    lg = lg >= 0.f ? lg : NEG_SLOPE * lg;
    float m = dec_f(mkey[d * H + hh]);
    atomicAdd(&ssum[d * H + hh], expf(lg - m));
}

// ---------------- edge pass 3: out[dst] += alpha * z[src] -------------------
// One wave per (edge, head); 64 channels -> float2 per lane.
__global__ __launch_bounds__(256)
void edge_aggr_kernel(const int* __restrict__ src, const int* __restrict__ dst,
                      const float* __restrict__ s_src, const float* __restrict__ s_dst,
                      const unsigned* __restrict__ mkey, const float* __restrict__ ssum,
                      const float* __restrict__ z, float* __restrict__ out,
                      int E, int H) {
    int lane = threadIdx.x & 31;
    int wid  = blockIdx.x * (blockDim.x >> 5) + (threadIdx.x >> 5);
    if (wid >= E * H) return;
    int e = wid / H, hh = wid - e * H;
    int s = src[e], d = dst[e];
    float lg = s_src[s * H + hh] + s_dst[d * H + hh];
    lg = lg >= 0.f ? lg : NEG_SLOPE * lg;
    float alpha = expf(lg - dec_f(mkey[d * H + hh])) / ssum[d * H + hh];
    const float2* zr = (const float2*)(z + (size_t)(s * H + hh) * 64);
    float2 v = zr[lane];
    float* orow = out + (size_t)(d * H + hh) * 64 + 2 * lane;
    atomicAdd(orow + 0, alpha * v.x);
    atomicAdd(orow + 1, alpha * v.y);
}

// ---------------- elementwise ELU (in place) --------------------------------
__global__ __launch_bounds__(256)
void elu_kernel(float* __restrict__ x, int n) {
    int i = blockIdx.x * blockDim.x + threadIdx.x;
    if (i >= n) return;
    float v = x[i];
    x[i] = v > 0.f ? v : expm1f(v);
}

// ===========================================================================
extern "C" void kernel_launch(void* const* d_in, const int* in_sizes, int n_in,
                              void* d_out, int out_size, void* d_ws, size_t ws_size,
                              hipStream_t stream) {
    const float* h   = (const float*)d_in[0];   // [N,128]
    const int*   src = (const int*)  d_in[1];   // [E]
    const int*   dst = (const int*)  d_in[2];   // [E]
    const float* W1  = (const float*)d_in[3];   // [4,128,64]
    const float* a1  = (const float*)d_in[4];   // [4,128]
    const float* W2  = (const float*)d_in[5];   // [256,64]
    const float* a2  = (const float*)d_in[6];   // [128]
    float* out = (float*)d_out;                 // [N,64]

    const int N = in_sizes[0] / IN_DIM;
    const int E = in_sizes[1];
    const int W1ROWS = HEADS * HID;             // 256

    // ---- workspace carve-up (256B aligned) ----
    char* ws = (char*)d_ws;
    size_t off = 0;
    auto carve = [&](size_t bytes) -> char* {
        char* p = ws + off;
        off = (off + bytes + 255) & ~(size_t)255;
        return p;
    };
    float*    z1     = (float*)   carve((size_t)N * W1ROWS * 4);  // [N,256] head-major
    float*    h1     = (float*)   carve((size_t)N * W1ROWS * 4);  // out1 then ELU'd
    float*    z2     = (float*)   carve((size_t)N * OUTD * 4);    // [N,64]
    float*    ssrc1  = (float*)   carve((size_t)N * HEADS * 4);
    float*    sdst1  = (float*)   carve((size_t)N * HEADS * 4);
    unsigned* mkey1  = (unsigned*)carve((size_t)N * HEADS * 4);
    float*    ssum1  = (float*)   carve((size_t)N * HEADS * 4);
    float*    ssrc2  = (float*)   carve((size_t)N * 4);
    float*    sdst2  = (float*)   carve((size_t)N * 4);
    unsigned* mkey2  = (unsigned*)carve((size_t)N * 4);
    float*    ssum2  = (float*)   carve((size_t)N * 4);
    (void)ws_size;

    const int BLK = 256;
    auto cdiv = [](long long a, long long b) { return (int)((a + b - 1) / b); };

    // ---- init accumulators ----
    hipMemsetAsync(ssum1, 0, (size_t)N * HEADS * 4, stream);
    hipMemsetAsync(h1,    0, (size_t)N * W1ROWS * 4, stream);
    hipMemsetAsync(ssum2, 0, (size_t)N * 4, stream);
    hipMemsetAsync(out,   0, (size_t)N * OUTD * 4, stream);
    init_keys_kernel<<<cdiv(N * HEADS, BLK), BLK, 0, stream>>>(mkey1, N * HEADS);
    init_keys_kernel<<<cdiv(N, BLK), BLK, 0, stream>>>(mkey2, N);

    // ---- layer 1: z1 = h @ W1 (all heads), via f32 WMMA ----
    {
        long long waves = (long long)cdiv(N, 16) * (W1ROWS / 16);
        gemm_wmma_f32<<<cdiv(waves * 32, BLK), BLK, 0, stream>>>(
            h, IN_DIM, IN_DIM, W1, IN_DIM * HID /*head stride*/, z1, W1ROWS, W1ROWS, N);
    }
    scores_kernel<<<cdiv((long long)N * HEADS, BLK), BLK, 0, stream>>>(
        z1, a1, ssrc1, sdst1, N, HEADS, HID);

    // ---- layer 1: edge softmax + aggregate ----
    edge_max_kernel<<<cdiv((long long)E * HEADS, BLK), BLK, 0, stream>>>(
        src, dst, ssrc1, sdst1, mkey1, E, HEADS);
    edge_sum_kernel<<<cdiv((long long)E * HEADS, BLK), BLK, 0, stream>>>(
        src, dst, ssrc1, sdst1, mkey1, ssum1, E, HEADS);
    edge_aggr_kernel<<<cdiv((long long)E * HEADS * 32, BLK), BLK, 0, stream>>>(
        src, dst, ssrc1, sdst1, mkey1, ssum1, z1, h1, E, HEADS);

    // ---- ELU ----
    elu_kernel<<<cdiv((long long)N * W1ROWS, BLK), BLK, 0, stream>>>(h1, N * W1ROWS);

    // ---- layer 2: z2 = h1 @ W2, via f32 WMMA ----
    {
        long long waves = (long long)cdiv(N, 16) * (OUTD / 16);
        gemm_wmma_f32<<<cdiv(waves * 32, BLK), BLK, 0, stream>>>(
            h1, W1ROWS, W1ROWS, W2, 0 /*single head*/, z2, OUTD, OUTD, N);
    }
    scores_kernel<<<cdiv(N, BLK), BLK, 0, stream>>>(z2, a2, ssrc2, sdst2, N, 1, OUTD);

    // ---- layer 2: edge softmax + aggregate into d_out ----
    edge_max_kernel<<<cdiv(E, BLK), BLK, 0, stream>>>(src, dst, ssrc2, sdst2, mkey2, E, 1);
    edge_sum_kernel<<<cdiv(E, BLK), BLK, 0, stream>>>(src, dst, ssrc2, sdst2, mkey2, ssum2, E, 1);
    edge_aggr_kernel<<<cdiv((long long)E * 32, BLK), BLK, 0, stream>>>(
        src, dst, ssrc2, sdst2, mkey2, ssum2, z2, out, E, 1);
}